// DAGNN_60928406061382
// MI455X (gfx1250) — compile-verified
//
#include <hip/hip_runtime.h>
#include <hip/hip_bf16.h>

typedef __attribute__((ext_vector_type(2))) float v2f;
typedef __attribute__((ext_vector_type(4))) float f4;
typedef __attribute__((ext_vector_type(8))) float v8f;

#define N_FEAT  512
#define N_HID   256
#define N_CLASS 40
#define K_HOPS  10
#define XSTR    (N_CLASS * (K_HOPS + 1))   // 440 floats per node in xs

// ---------------------------------------------------------------------------
// zero workspace (xs) — re-run every call so graph replay stays deterministic
// ---------------------------------------------------------------------------
__global__ __launch_bounds__(256) void zero_kernel(f4* __restrict__ p, long long n4) {
    long long i = (long long)blockIdx.x * 256 + threadIdx.x;
    if (i < n4) {
        f4 z = {};
        p[i] = z;
    }
}

// ---------------------------------------------------------------------------
// Fused MLP: xs[:,0,:] = relu(feat@W1+b1)@W2+b2 using V_WMMA_F32_16X16X4_F32
// Block: 256 thr = 8 waves, owns 32 node rows.
//  - feature tile staged with GLOBAL_LOAD_ASYNC_TO_LDS_B128 (ASYNCcnt path)
//  - W1 staged k-pair-interleaved so each B fragment is one ds_load_b64
//  - z kept in LDS; GEMM2 (waves 0..5) computes the 32x48(pad of 40) tile
// ---------------------------------------------------------------------------
__global__ __launch_bounds__(256) void mlp_fused(
    const float* __restrict__ feat, const float* __restrict__ W1,
    const float* __restrict__ b1,   const float* __restrict__ W2,
    const float* __restrict__ b2,   float* __restrict__ xs, int n)
{
    __shared__ float Asm[32][36];              // feature tile (pad 36: conflict-free col reads)
    __shared__ float Bsm2[16][N_HID + 16][2];  // W1 band, k-pair interleaved: [k>>1][col][k&1]
    __shared__ float Zsm[32][N_HID + 4];       // z tile (pad 260: conflict-free col reads)

    const int tid  = threadIdx.x;
    const int wave = tid >> 5;
    const int lane = tid & 31;
    const int half = lane >> 4;            // which 16-lane half of the wave
    const int m16  = lane & 15;
    const int n0   = blockIdx.x * 32;

    const int rs = wave >> 2;              // row strip 0..1
    const int cg = wave & 3;               // col group 0..3 (64 cols each)

    v8f acc[4] = {};

    for (int kb = 0; kb < N_FEAT; kb += 32) {
        // ---- stage 32x32 feature tile: async global->LDS, one b128 per thread
        {
            int r  = tid >> 3;
            int c4 = (tid & 7) << 2;
            int gr = n0 + r; if (gr >= n) gr = n - 1;
            unsigned int lds_dst =
                (unsigned int)(unsigned long long)(uintptr_t)(&Asm[r][c4]);
            unsigned long long gaddr =
                (unsigned long long)(uintptr_t)(feat + (long long)gr * N_FEAT + kb + c4);
            asm volatile("global_load_async_to_lds_b128 %0, %1, off"
                         :: "v"(lds_dst), "v"(gaddr) : "memory");
        }
        // ---- stage 32x256 W1 band, k-pair interleaved (8 packed float4 per thread)
        #pragma unroll
        for (int i = 0; i < 8; ++i) {
            int idx = tid + i * 256;
            int k2  = idx >> 7;             // 0..15
            int c   = (idx & 127) << 1;     // 0..254 step 2
            v2f lo = *(const v2f*)(W1 + (long long)(kb + 2 * k2)     * N_HID + c);
            v2f hi = *(const v2f*)(W1 + (long long)(kb + 2 * k2 + 1) * N_HID + c);
            f4 packed = {lo.x, hi.x, lo.y, hi.y};
            *(f4*)(&Bsm2[k2][c][0]) = packed;
        }
        // wave-level completion of this wave's async LDS writes, then block barrier
        asm volatile("s_wait_asynccnt 0" ::: "memory");
        __syncthreads();

        #pragma unroll
        for (int k4 = 0; k4 < 32; k4 += 4) {
            const int kk = k4 + 2 * half;   // A frag: vgpr=k&1, lane-half=k>>1
            v2f a = *(const v2f*)(&Asm[rs * 16 + m16][kk]);
            #pragma unroll
            for (int j = 0; j < 4; ++j) {
                const int col = cg * 64 + j * 16 + m16;
                v2f b = *(const v2f*)(&Bsm2[(k4 >> 1) + half][col][0]);
                acc[j] = __builtin_amdgcn_wmma_f32_16x16x4_f32(
                    false, a, false, b, (short)0, acc[j], false, false);
            }
        }
        __syncthreads();
    }

    // z = relu(acc + b1) -> LDS (C/D layout: M = r + 8*half, N = m16)
    #pragma unroll
    for (int j = 0; j < 4; ++j) {
        const int col = cg * 64 + j * 16 + m16;
        const float bv = b1[col];
        #pragma unroll
        for (int r = 0; r < 8; ++r) {
            const int M = rs * 16 + r + 8 * half;
            float v = acc[j][r] + bv;
            Zsm[M][col] = v > 0.f ? v : 0.f;
        }
    }
    __syncthreads();

    // GEMM2: 32 rows x 40 cols (padded to 48): 2 row strips x 3 col tiles = waves 0..5
    if (wave < 6) {
        const int rs2 = wave / 3;
        const int jc  = wave % 3;
        const int col = jc * 16 + m16;
        const bool cv = (col < N_CLASS);
        v8f acc2 = {};
        for (int kk = 0; kk < N_HID; kk += 4) {
            const int k = kk + 2 * half;
            v2f a = *(const v2f*)(&Zsm[rs2 * 16 + m16][k]);
            v2f b;
            b.x = cv ? W2[(long long)k * N_CLASS + col] : 0.f;
            b.y = cv ? W2[(long long)(k + 1) * N_CLASS + col] : 0.f;
            acc2 = __builtin_amdgcn_wmma_f32_16x16x4_f32(
                false, a, false, b, (short)0, acc2, false, false);
        }
        if (cv) {
            const float b2v = b2[col];
            #pragma unroll
            for (int r = 0; r < 8; ++r) {
                const int node = n0 + rs2 * 16 + r + 8 * half;
                if (node < n)
                    xs[(long long)node * XSTR + col] = acc2[r] + b2v;
            }
        }
    }
}

// ---------------------------------------------------------------------------
// One propagation hop: xs_out[row,c] += w_e * xs_in[col,c]
// One thread per (edge, channel); atomics resolve in L2 (x slice ~17.6MB).
// ---------------------------------------------------------------------------
__global__ __launch_bounds__(256) void spmm_kernel(
    const int* __restrict__ erow, const int* __restrict__ ecol,
    const float* __restrict__ ew, const float* __restrict__ xin,
    float* __restrict__ xout, unsigned int total)
{
    unsigned int idx = blockIdx.x * 256u + threadIdx.x;
    if (idx >= total) return;
    unsigned int e = idx / (unsigned)N_CLASS;
    unsigned int c = idx - e * (unsigned)N_CLASS;
    int   r  = erow[e];
    int   cl = ecol[e];
    float w  = ew[e];
    float v  = xin[(long long)cl * XSTR + c];
    atomicAdd(xout + (long long)r * XSTR + c, w * v);
}

// ---------------------------------------------------------------------------
// Attention over hops + log_softmax. One wave per node; lane holds channels
// c=lane and c=lane+32 (lanes 0..7).
// ---------------------------------------------------------------------------
__global__ __launch_bounds__(256) void attn_kernel(
    const float* __restrict__ xs, const float* __restrict__ Wa,
    const float* __restrict__ ba, float* __restrict__ out, int n)
{
    const int wave = threadIdx.x >> 5;
    const int lane = threadIdx.x & 31;
    const int node = blockIdx.x * 8 + wave;
    if (node >= n) return;

    const float* xn = xs + (long long)node * XSTR;
    const int  c1 = lane;
    const int  c2 = lane + 32;
    const bool has2 = (c2 < N_CLASS);

    const float wa1 = Wa[c1];
    const float wa2 = has2 ? Wa[c2] : 0.f;
    const float bav = ba[0];

    float x1[K_HOPS + 1], x2[K_HOPS + 1], sc[K_HOPS + 1];
    #pragma unroll
    for (int k = 0; k <= K_HOPS; ++k) {
        x1[k] = xn[k * N_CLASS + c1];
        x2[k] = has2 ? xn[k * N_CLASS + c2] : 0.f;
    }
    #pragma unroll
    for (int k = 0; k <= K_HOPS; ++k) {
        float p = x1[k] * wa1 + x2[k] * wa2;
        #pragma unroll
        for (int off = 16; off > 0; off >>= 1)
            p += __shfl_xor(p, off, 32);
        sc[k] = 1.f / (1.f + expf(-(p + bav)));
    }
    float o1 = 0.f, o2 = 0.f;
    #pragma unroll
    for (int k = 0; k <= K_HOPS; ++k) {
        o1 += sc[k] * x1[k];
        o2 += sc[k] * x2[k];
    }
    // log_softmax over 40 channels spread across the wave
    float mx = has2 ? fmaxf(o1, o2) : o1;
    #pragma unroll
    for (int off = 16; off > 0; off >>= 1)
        mx = fmaxf(mx, __shfl_xor(mx, off, 32));
    float se = expf(o1 - mx) + (has2 ? expf(o2 - mx) : 0.f);
    #pragma unroll
    for (int off = 16; off > 0; off >>= 1)
        se += __shfl_xor(se, off, 32);
    const float lse = mx + logf(se);

    out[(long long)node * N_CLASS + c1] = o1 - lse;
    if (has2) out[(long long)node * N_CLASS + c2] = o2 - lse;
}

// ---------------------------------------------------------------------------
extern "C" void kernel_launch(void* const* d_in, const int* in_sizes, int n_in,
                              void* d_out, int out_size, void* d_ws, size_t ws_size,
                              hipStream_t stream) {
    const float* feature = (const float*)d_in[0];
    const int*   erow    = (const int*)  d_in[1];
    const int*   ecol    = (const int*)  d_in[2];
    const float* ew      = (const float*)d_in[3];
    const float* W1      = (const float*)d_in[4];
    const float* b1      = (const float*)d_in[5];
    const float* W2      = (const float*)d_in[6];
    const float* b2      = (const float*)d_in[7];
    const float* Wa      = (const float*)d_in[8];
    const float* ba      = (const float*)d_in[9];
    float* out = (float*)d_out;

    const int n  = in_sizes[0] / N_FEAT;   // nodes
    const int ne = in_sizes[1];            // edges
    float* xs = (float*)d_ws;              // [n][K_HOPS+1][N_CLASS] fp32

    // 1) zero xs (hop buffers accumulate with atomics)
    const long long n4 = (long long)n * XSTR / 4;
    zero_kernel<<<(unsigned)((n4 + 255) / 256), 256, 0, stream>>>((f4*)xs, n4);

    // 2) fused MLP -> hop 0
    mlp_fused<<<(n + 31) / 32, 256, 0, stream>>>(feature, W1, b1, W2, b2, xs, n);

    // 3) K sparse propagation hops
    const unsigned int total  = (unsigned int)ne * (unsigned int)N_CLASS;
    const unsigned int blocks = (total + 255u) / 256u;
    for (int k = 0; k < K_HOPS; ++k)
        spmm_kernel<<<blocks, 256, 0, stream>>>(erow, ecol, ew,
                                                xs + k * N_CLASS,
                                                xs + (k + 1) * N_CLASS, total);

    // 4) hop attention + log_softmax
    attn_kernel<<<(n + 7) / 8, 256, 0, stream>>>(xs, Wa, ba, out, n);
}